// MaceBlock_36146444763352
// MI455X (gfx1250) — compile-verified
//
#include <hip/hip_runtime.h>
#include <math.h>

// ---------------------------------------------------------------------------
// MACE-like equivariant GNN block for MI455X (gfx1250, wave32, WMMA).
// Pipeline:
//   K1 edge_geom       : Y(E,16), rb(E,8)
//   gemm16_wmma        : up0 = embed[spec] @ lu0           (WMMA f32 16x16x4)
//   msg_scatter<0>     : fused radial-MLP GEMM (WMMA) + tensor-product msg
//                        + atomicAdd scatter into agg (L2-resident, 41MB)
//   gemm16_wmma x16    : f = (agg/16) @ ld0[l(lm)]         (WMMA)
//   node_sel_prod      : sel_w / prod_w0 / read0 / res_w   (VALU, per-node)
//   gemm16_wmma x4     : up1 from f1 @ lu1                  (WMMA)
//   msg_scatter<1>     : second interaction (WMMA + scatter)
//   gemm16_wmma x16    : g = (agg/16) @ ld1[l(lm)]         (WMMA)
//   node_final         : invariants + prod_w1 + res + MLP readout
// ---------------------------------------------------------------------------

typedef float v2f __attribute__((ext_vector_type(2)));
typedef float v8f __attribute__((ext_vector_type(8)));

__device__ __forceinline__ float silu_f(float x) { return x / (1.0f + expf(-x)); }
__device__ __forceinline__ int l_of_lm(int lm) {
  return (lm == 0) ? 0 : ((lm < 4) ? 1 : ((lm < 9) ? 2 : 3));
}

// D(16x16,f32) += A(16x4,f32) x B(4x16,f32), accumulated over K in steps of 4.
// Lane layout (wave32): lanes 0-15 hold {K,K+1} for row=lane; lanes 16-31 hold
// {K+2,K+3} for row=lane-16. D: VGPR j = rows j (lanes 0-15) / j+8 (16-31).
__device__ __forceinline__ v8f wmma_f32(v2f a, v2f b, v8f c) {
  return __builtin_amdgcn_wmma_f32_16x16x4_f32(false, a, false, b, (short)0, c,
                                               false, false);
}

// ---------------------------------------------------------------------------
// K1: per-edge geometry: spherical harmonics Y (16) + radial embedding rb (8)
// ---------------------------------------------------------------------------
__global__ __launch_bounds__(256)
void edge_geom_kernel(const float* __restrict__ ev, float* __restrict__ Yg,
                      float* __restrict__ rbg, int E) {
  int e = blockIdx.x * 256 + threadIdx.x;
  if (e >= E) return;
  float x = ev[(size_t)e * 3 + 0];
  float y = ev[(size_t)e * 3 + 1];
  float z = ev[(size_t)e * 3 + 2];
  float r = sqrtf(x * x + y * y + z * z);
  float inv = 1.0f / (r + 1e-8f);
  float ux = x * inv, uy = y * inv, uz = z * inv;
  float x2 = ux * ux, y2 = uy * uy, z2 = uz * uz;
  float Ye[16];
  Ye[0] = 1.0f;
  Ye[1] = 1.7320508f * ux;
  Ye[2] = 1.7320508f * uy;
  Ye[3] = 1.7320508f * uz;
  Ye[4] = 3.8729833f * ux * uy;
  Ye[5] = 3.8729833f * uy * uz;
  Ye[6] = 1.118034f * (3.0f * z2 - 1.0f);
  Ye[7] = 3.8729833f * ux * uz;
  Ye[8] = 1.9364917f * (x2 - y2);
  Ye[9] = 2.09165f * uy * (3.0f * x2 - y2);
  Ye[10] = 10.246951f * ux * uy * uz;
  Ye[11] = 1.6201852f * uy * (5.0f * z2 - 1.0f);
  Ye[12] = 1.3228757f * uz * (5.0f * z2 - 3.0f);
  Ye[13] = 1.6201852f * ux * (5.0f * z2 - 1.0f);
  Ye[14] = 5.1234753f * uz * (x2 - y2);
  Ye[15] = 2.09165f * ux * (x2 - 3.0f * y2);
#pragma unroll
  for (int i = 0; i < 16; ++i) Yg[(size_t)e * 16 + i] = Ye[i];

  // radial: sqrt(2/5) * sin(n*pi*r/5)/(r+1e-8) * polynomial envelope (p=5)
  float t = fminf(fmaxf(r * 0.2f, 0.0f), 1.0f);
  float t2 = t * t, t4 = t2 * t2, t5 = t4 * t, t6 = t5 * t, t7 = t6 * t;
  float env = 1.0f - 21.0f * t5 + 35.0f * t6 - 15.0f * t7;
  float cf = 0.63245553f * inv * env;  // sqrt(2/5)
  float w = 0.62831853f * r;           // pi*r/5
#pragma unroll
  for (int k = 1; k <= 8; ++k)
    rbg[(size_t)e * 8 + (k - 1)] = cf * sinf((float)k * w);
}

// ---------------------------------------------------------------------------
// Generic 16-row WMMA GEMM: D[n, col, m] = a_scale * A[n(,m)] (16x64) @ B (64x64)
// b_mode: 0 -> B[0];  1 -> B[m>0];  2 -> B[l_of_lm(m)]
// Block = 128 threads (4 waves, one 16-col tile each). Grid = (rowTiles, M).
// ---------------------------------------------------------------------------
__global__ __launch_bounds__(128)
void gemm16_wmma(const float* __restrict__ A, const int* __restrict__ gatherIdx,
                 long a_row_stride, int a_col_stride, long a_slice_off,
                 float a_scale, const float* __restrict__ B, int b_mode,
                 float* __restrict__ D, long d_row_stride, int d_col_stride,
                 long d_slice_off, int Nrows) {
  __shared__ float As[16][68];
  __shared__ float Bs[64 * 64];
  const int tid = threadIdx.x;
  const int n0 = blockIdx.x << 4;
  const int m = blockIdx.y;
  const int bidx = (b_mode == 0) ? 0 : ((b_mode == 1) ? (m > 0 ? 1 : 0) : l_of_lm(m));
  const float* Bm = B + (size_t)bidx * 4096;

  for (int idx = tid; idx < 1024; idx += 128) {
    int rr = idx >> 6, c = idx & 63;
    int n = n0 + rr;
    if (n >= Nrows) n = Nrows - 1;
    long arow = gatherIdx ? (long)gatherIdx[n] : (long)n;
    As[rr][c] = a_scale * A[arow * a_row_stride + (long)c * a_col_stride +
                            (long)m * a_slice_off];
  }
  for (int idx = tid; idx < 4096; idx += 128) Bs[idx] = Bm[idx];
  __syncthreads();

  const int wave = tid >> 5, lane = tid & 31, half = lane >> 4, r = lane & 15;
  const int col0 = wave * 16;
  v8f acc = {0.f, 0.f, 0.f, 0.f, 0.f, 0.f, 0.f, 0.f};
#pragma unroll
  for (int k4 = 0; k4 < 16; ++k4) {
    const int kb = k4 * 4 + half * 2;
    v2f a = {As[r][kb], As[r][kb + 1]};
    v2f b = {Bs[kb * 64 + col0 + r], Bs[(kb + 1) * 64 + col0 + r]};
    acc = wmma_f32(a, b, acc);
  }
#pragma unroll
  for (int j = 0; j < 8; ++j) {
    int row = n0 + j + 8 * half;
    if (row < Nrows)
      D[(long)row * d_row_stride + (long)(col0 + r) * d_col_stride +
        (long)m * d_slice_off] = acc[j];
  }
}

// ---------------------------------------------------------------------------
// Fused per-edge kernel: radial MLP (WMMA) + tensor-product message + scatter.
// Block = 256 threads (8 waves) per 16-edge tile.
//   Phase 1: H(16x64) = silu(rb @ rw1)  -> LDS
//   Phase 2: R(16x768) = H @ rw2 via v_wmma_f32_16x16x4_f32 (wave w: cols 96w..)
//   Phase 3: msg[e,c,lm] from R/hs/Y, atomicAdd into agg[recv] (L2-resident)
// R layout per edge: col = c*12 + j  (j: 0..3=R0, 4..7=R1, 8..11=R2)
// ---------------------------------------------------------------------------
template <int INTER>
__global__ __launch_bounds__(256)
void msg_scatter_kernel(const float* __restrict__ rbg, const float* __restrict__ Yg,
                        const float* __restrict__ rw1, const float* __restrict__ rw2,
                        const float* __restrict__ up, const int* __restrict__ senders,
                        const int* __restrict__ receivers, float* __restrict__ agg,
                        int E) {
  __shared__ float Hs[16][68];
  __shared__ float Rs[16][772];
  __shared__ float Ys[16][16];
  __shared__ int snd[16];
  __shared__ int rcv[16];

  const int tid = threadIdx.x;
  const int e0 = blockIdx.x << 4;

  {
    int e = tid >> 4, lm = tid & 15;
    int eg = e0 + e;
    if (eg >= E) eg = E - 1;
    Ys[e][lm] = Yg[(size_t)eg * 16 + lm];
  }
  if (tid < 16) {
    int eg = e0 + tid;
    if (eg >= E) eg = E - 1;
    snd[tid] = senders[eg];
    rcv[tid] = receivers[eg];
  }

  // Phase 1: hidden layer
  for (int idx = tid; idx < 16 * 64; idx += 256) {
    int e = idx >> 6, j = idx & 63;
    int eg = e0 + e;
    if (eg >= E) eg = E - 1;
    const float* rbe = rbg + (size_t)eg * 8;
    float acc = 0.0f;
#pragma unroll
    for (int k = 0; k < 8; ++k) acc += rbe[k] * rw1[k * 64 + j];
    Hs[e][j] = silu_f(acc);
  }
  __syncthreads();

  // Phase 2: R = H @ rw2 (rw2 is 196KB, L2-resident broadcast)
  const int wave = tid >> 5, lane = tid & 31, half = lane >> 4, r = lane & 15;
  for (int ct = 0; ct < 6; ++ct) {
    const int col0 = wave * 96 + ct * 16;
    v8f acc = {0.f, 0.f, 0.f, 0.f, 0.f, 0.f, 0.f, 0.f};
#pragma unroll
    for (int k4 = 0; k4 < 16; ++k4) {
      const int kb = k4 * 4 + half * 2;
      v2f a = {Hs[r][kb], Hs[r][kb + 1]};
      v2f b = {rw2[(size_t)kb * 768 + col0 + r],
               rw2[(size_t)(kb + 1) * 768 + col0 + r]};
      acc = wmma_f32(a, b, acc);
    }
#pragma unroll
    for (int j = 0; j < 8; ++j) Rs[j + 8 * half][col0 + r] = acc[j];
  }
  __syncthreads();

  // Phase 3: messages + scatter-add
  for (int idx = tid; idx < 16 * 64; idx += 256) {
    int e = idx >> 6, c = idx & 63;
    if (e0 + e >= E) continue;
    const float* Rec = &Rs[e][c * 12];
    const int s = snd[e];
    float* aggp = agg + (size_t)rcv[e] * 1024 + c * 16;
    if (INTER == 0) {
      const float hs0 = up[(size_t)s * 64 + c];
#pragma unroll
      for (int lm = 0; lm < 16; ++lm) {
        float mv = Rec[l_of_lm(lm)] * hs0 * Ys[e][lm];
        atomicAdd(aggp + lm, mv);
      }
    } else {
      const float hs0 = up[(size_t)s * 256 + c * 4 + 0];
      const float hs1 = up[(size_t)s * 256 + c * 4 + 1];
      const float hs2 = up[(size_t)s * 256 + c * 4 + 2];
      const float hs3 = up[(size_t)s * 256 + c * 4 + 3];
      const float R1v = Rec[5];  // R1[l=1]
      const float R2v = Rec[9];  // R2[l=1]
      float hsm1[3] = {hs1, hs2, hs3};
#pragma unroll
      for (int lm = 0; lm < 16; ++lm) {
        float mv = Rec[l_of_lm(lm)] * hs0 * Ys[e][lm];
        if (lm >= 1 && lm < 4) mv += R1v * hsm1[lm - 1] * Ys[e][0];
        if (lm == 0)
          mv += R2v * (hs1 * Ys[e][1] + hs2 * Ys[e][2] + hs3 * Ys[e][3]);
        atomicAdd(aggp + lm, mv);
      }
    }
  }
}

// ---------------------------------------------------------------------------
// K5: species-conditioned mixing after interaction 0 (one block per node).
// fsel[d,lm] = sum_c f[c,lm]*sel_w[sp,l(lm),c,d]; fac = 1+s+s^2 (s=fsel[:,0]);
// f1[d,m] = sum_c fsel[c,m]*fac[c]*prod_w0[sp,c,d]; out0; res.
// ---------------------------------------------------------------------------
__global__ __launch_bounds__(256)
void node_sel_prod_kernel(const float* __restrict__ f, const int* __restrict__ species,
                          const float* __restrict__ sel_w,
                          const float* __restrict__ prod_w0,
                          const float* __restrict__ res_w,
                          const float* __restrict__ read0, float* __restrict__ f1,
                          float* __restrict__ resb, float* __restrict__ out) {
  __shared__ float fs[64][16];
  __shared__ float gsel[64][17];
  __shared__ float fac[64];
  __shared__ float f1s[64][4];
  const int tid = threadIdx.x;
  const int n = blockIdx.x;
  const int sp = species[n];

  for (int idx = tid; idx < 1024; idx += 256)
    fs[idx >> 4][idx & 15] = f[(size_t)n * 1024 + idx];
  __syncthreads();

#pragma unroll
  for (int ii = 0; ii < 4; ++ii) {
    int idx = tid + 256 * ii;
    int lm = idx >> 6, d = idx & 63;
    const float* W = sel_w + ((size_t)sp * 4 + l_of_lm(lm)) * 4096;
    float acc = 0.0f;
    for (int c = 0; c < 64; ++c) acc += fs[c][lm] * W[c * 64 + d];
    gsel[d][lm] = acc;
  }
  __syncthreads();

  if (tid < 64) {
    float s = gsel[tid][0];
    fac[tid] = 1.0f + s + s * s;
  }
  __syncthreads();

  {
    int d = tid & 63, mm = tid >> 6;
    const float* W = prod_w0 + (size_t)sp * 4096;
    float acc = 0.0f;
    for (int c = 0; c < 64; ++c) acc += gsel[c][mm] * fac[c] * W[c * 64 + d];
    f1s[d][mm] = acc;
    f1[(size_t)n * 256 + d * 4 + mm] = acc;
  }
  __syncthreads();

  if (tid == 0) {
    float acc = 0.0f;
    for (int d = 0; d < 64; ++d) acc += f1s[d][0] * read0[d];
    out[(size_t)n * 2 + 0] = acc;
  }
  if (tid < 64) {
    const float* W = res_w + (size_t)sp * 4096;
    float acc = 0.0f;
    for (int c = 0; c < 64; ++c) acc += f1s[c][0] * W[c * 64 + tid];
    resb[(size_t)n * 64 + tid] = acc;
  }
}

// ---------------------------------------------------------------------------
// K6: invariants + prod_w1 + residual + 2-layer MLP readout (one block/node).
// ---------------------------------------------------------------------------
__global__ __launch_bounds__(64)
void node_final_kernel(const float* __restrict__ g, const int* __restrict__ species,
                       const float* __restrict__ prod_w1,
                       const float* __restrict__ resb,
                       const float* __restrict__ mlp_w1,
                       const float* __restrict__ mlp_w2, float* __restrict__ out) {
  __shared__ float q[64];
  __shared__ float f2[64];
  __shared__ float tmid[32];
  const int tid = threadIdx.x;
  const int n = blockIdx.x;
  const int sp = species[n];

  {
    const float* grow = g + (size_t)n * 1024 + tid * 16;
    float inv = grow[0];
    float ss = 0.0f;
#pragma unroll
    for (int lm = 1; lm < 16; ++lm) ss += grow[lm] * grow[lm];
    inv += ss;
    q[tid] = inv + inv * inv;
  }
  __syncthreads();
  {
    const float* W = prod_w1 + (size_t)sp * 4096;
    float acc = 0.0f;
    for (int c = 0; c < 64; ++c) acc += q[c] * W[c * 64 + tid];
    f2[tid] = acc + resb[(size_t)n * 64 + tid];
  }
  __syncthreads();
  if (tid < 32) {
    float acc = 0.0f;
    for (int d = 0; d < 64; ++d) acc += f2[d] * mlp_w1[d * 32 + tid];
    tmid[tid] = silu_f(acc);
  }
  __syncthreads();
  if (tid == 0) {
    float acc = 0.0f;
#pragma unroll
    for (int j = 0; j < 32; ++j) acc += tmid[j] * mlp_w2[j];
    out[(size_t)n * 2 + 1] = acc;
  }
}

// ---------------------------------------------------------------------------
extern "C" void kernel_launch(void* const* d_in, const int* in_sizes, int n_in,
                              void* d_out, int out_size, void* d_ws, size_t ws_size,
                              hipStream_t stream) {
  const float* edge_vectors = (const float*)d_in[0];
  const float* node_embed = (const float*)d_in[1];
  const float* rw1 = (const float*)d_in[2];   // (2,8,64)
  const float* rw2 = (const float*)d_in[3];   // (2,64,768)
  const float* lu0 = (const float*)d_in[4];   // (1,64,64)
  const float* lu1 = (const float*)d_in[5];   // (2,64,64)
  const float* ld = (const float*)d_in[6];    // (2,4,64,64)
  const float* sel_w = (const float*)d_in[7]; // (10,4,64,64)
  const float* prod_w0 = (const float*)d_in[8];
  const float* prod_w1 = (const float*)d_in[9];
  const float* res_w = (const float*)d_in[10];
  const float* read0 = (const float*)d_in[11];
  const float* mlp_w1 = (const float*)d_in[12];
  const float* mlp_w2 = (const float*)d_in[13];
  const int* species = (const int*)d_in[14];
  const int* senders = (const int*)d_in[15];
  const int* receivers = (const int*)d_in[16];
  float* out = (float*)d_out;

  const int E = in_sizes[0] / 3;
  const int N = in_sizes[14];

  float* w = (float*)d_ws;
  float* Yb = w;   w += (size_t)E * 16;
  float* rbb = w;  w += (size_t)E * 8;
  float* up0 = w;  w += (size_t)N * 64;
  float* up1 = w;  w += (size_t)N * 256;
  float* agg = w;  w += (size_t)N * 1024;
  float* fbuf = w; w += (size_t)N * 1024;
  float* f1b = w;  w += (size_t)N * 256;
  float* resb = w; w += (size_t)N * 64;
  (void)ws_size; (void)n_in; (void)out_size;

  const int ET = (E + 15) / 16;
  const int NT = (N + 15) / 16;

  // K1: per-edge geometry
  edge_geom_kernel<<<(E + 255) / 256, 256, 0, stream>>>(edge_vectors, Yb, rbb, E);

  // Interaction 0
  hipMemsetAsync(agg, 0, (size_t)N * 1024 * sizeof(float), stream);
  // up0 = node_embed[species] @ lu0
  gemm16_wmma<<<dim3(NT, 1), 128, 0, stream>>>(node_embed, species, 64L, 1, 0L,
                                               1.0f, lu0, 0, up0, 64L, 1, 0L, N);
  msg_scatter_kernel<0><<<ET, 256, 0, stream>>>(rbb, Yb, rw1, rw2, up0, senders,
                                                receivers, agg, E);
  // f = (agg / AVG_NEI) @ ld[0][l(lm)]  for lm = 0..15
  gemm16_wmma<<<dim3(NT, 16), 128, 0, stream>>>(agg, nullptr, 1024L, 16, 1L,
                                                1.0f / 16.0f, ld, 2, fbuf, 1024L,
                                                16, 1L, N);
  // K5: sel / prod / readout0 / residual
  node_sel_prod_kernel<<<N, 256, 0, stream>>>(fbuf, species, sel_w, prod_w0,
                                              res_w, read0, f1b, resb, out);

  // Interaction 1
  hipMemsetAsync(agg, 0, (size_t)N * 1024 * sizeof(float), stream);
  // up1[:, :, m]: m=0 -> lu1[0], m=1..3 -> lu1[1]
  gemm16_wmma<<<dim3(NT, 4), 128, 0, stream>>>(f1b, nullptr, 256L, 4, 1L, 1.0f,
                                               lu1, 1, up1, 256L, 4, 1L, N);
  msg_scatter_kernel<1><<<ET, 256, 0, stream>>>(rbb, Yb, rw1 + 512, rw2 + 49152,
                                                up1, senders, receivers, agg, E);
  // g = (agg / AVG_NEI) @ ld[1][l(lm)]
  gemm16_wmma<<<dim3(NT, 16), 128, 0, stream>>>(agg, nullptr, 1024L, 16, 1L,
                                                1.0f / 16.0f, ld + 4 * 4096, 2,
                                                fbuf, 1024L, 16, 1L, N);
  // K6: invariants + readout1
  node_final_kernel<<<N, 64, 0, stream>>>(fbuf, species, prod_w1, resb, mlp_w1,
                                          mlp_w2, out);
}